// BlockCrossAttn_77678778515473
// MI455X (gfx1250) — compile-verified
//
#include <hip/hip_runtime.h>

typedef float v2f __attribute__((ext_vector_type(2)));
typedef float v8f __attribute__((ext_vector_type(8)));

#define BSZ    256   // attention block size (and workgroup size)
#define NBATCH 32    // N
#define WPB    8     // waves per workgroup (wave32)

__device__ __forceinline__ float read_lane_f(float v, int srclane) {
  return __int_as_float(__builtin_amdgcn_readlane(__float_as_int(v), srclane));
}

__device__ __forceinline__ float bperm_f(float v, int srclane) {
  return __int_as_float(__builtin_amdgcn_ds_bpermute(srclane << 2, __float_as_int(v)));
}

__device__ __forceinline__ float exp2_fast(float x) {
  return __builtin_amdgcn_exp2f(x);   // v_exp_f32
}

// Extract denominator (C col 0 -> lanes 0/16) and numerator (C col 1 -> lanes 1/17)
// for one 16-row q-group and store the projected outputs.
#define EMIT_OUT(C, GG)                                                          \
  {                                                                              \
    const size_t obase = (size_t)(blk * BSZ + (GG) * 16) * NBATCH + (size_t)n;   \
    _Pragma("unroll")                                                            \
    for (int r = 0; r < 8; ++r) {                                                \
      const float D0 = read_lane_f((C)[r], 0);                                   \
      const float N0 = read_lane_f((C)[r], 1);                                   \
      const float D1 = read_lane_f((C)[r], 16);                                  \
      const float N1 = read_lane_f((C)[r], 17);                                  \
      if (lane == r)     out[obase + (size_t)r * NBATCH]       = __fmaf_rn(wo, N0 / D0, bo); \
      if (lane == r + 8) out[obase + (size_t)(r + 8) * NBATCH] = __fmaf_rn(wo, N1 / D1, bo); \
    }                                                                            \
  }

__global__ __launch_bounds__(BSZ) void block_attn_kernel(
    const float* __restrict__ query,
    const float* __restrict__ key,
    const float* __restrict__ value,
    const float* __restrict__ in_w,   // [3]: wq, wk, wv
    const float* __restrict__ in_b,   // [3]
    const float* __restrict__ opw,    // [1]
    const float* __restrict__ opb,    // [1]
    float* __restrict__ out)
{
  __shared__ __align__(16) float qbuf[WPB][BSZ];
  __shared__ __align__(16) float kbuf[WPB][BSZ];
  __shared__ __align__(16) float vbuf[WPB][BSZ];

  const int tid  = threadIdx.x;
  const int wave = tid >> 5;
  const int lane = tid & 31;

  const int tile = blockIdx.x * WPB + wave;   // 0..4095
  const int blk  = tile >> 5;                 // sequence block 0..127
  const int n    = tile & 31;                 // batch index 0..31

  const float wq = in_w[0], wk = in_w[1], wv = in_w[2];
  const float bq = in_b[0], bk = in_b[1], bv = in_b[2];
  const float wo = opw[0],  bo = opb[0];

  // element (l, n) at l*NBATCH + n ; l = blk*BSZ + i
  const size_t g0 = (size_t)(blk * BSZ) * NBATCH + (size_t)n;

  // Stage projected q', k', v' for this tile into LDS; track k' min/max.
  float kmax = -3.402823466e38f, kmin = 3.402823466e38f;
  #pragma unroll
  for (int j = 0; j < 8; ++j) {
    const int i = lane + 32 * j;
    const size_t idx = g0 + (size_t)i * NBATCH;
    qbuf[wave][i] = __fmaf_rn(query[idx], wq, bq);       // scale = 1/sqrt(1)
    const float kv = __fmaf_rn(key[idx], wk, bk);
    kbuf[wave][i] = kv;
    kmax = fmaxf(kmax, kv);
    kmin = fminf(kmin, kv);
    vbuf[wave][i] = __fmaf_rn(value[idx], wv, bv);
  }
  // Cross-lane (wave32) max/min reduction via ds_bpermute butterfly.
  #pragma unroll
  for (int s = 16; s >= 1; s >>= 1) {
    kmax = fmaxf(kmax, bperm_f(kmax, lane ^ s));
    kmin = fminf(kmin, bperm_f(kmin, lane ^ s));
  }
  __syncthreads();

  const float* kl = kbuf[wave];
  const float* vl = vbuf[wave];

  // WMMA f32 16x16x4 operand mapping (wave32):
  //   A (16x4): lane half h = lane>>4, row m = lane&15, vgpr j <-> K = 2h+j
  //   B (4x16): lane half h picks K rows {2h, 2h+1}, column = lane&15
  //   C (16x16): vgpr r, lanes 0-15 -> M=r, N=lane ; lanes 16-31 -> M=8+r
  const int m   = lane & 15;
  const int h   = lane >> 4;
  const int col = lane & 15;

  // Lane-constant selectors: B[k][0] = 1 (row sum), B[k][1] = v'[k], rest 0.
  const float sel_v = (col == 1) ? 1.0f : 0.0f;
  const float sel_1 = (col == 0) ? 1.0f : 0.0f;
  const float LOG2E = 1.4426950408889634f;

  for (int g = 0; g < 16; g += 4) {
    // Per-row scaled logits: exp(q*k - mrow) = exp2(qs*k + nm)
    float qs0, qs1, qs2, qs3, nm0, nm1, nm2, nm3;
    {
      const float q0 = qbuf[wave][(g + 0) * 16 + m];
      const float q1 = qbuf[wave][(g + 1) * 16 + m];
      const float q2 = qbuf[wave][(g + 2) * 16 + m];
      const float q3 = qbuf[wave][(g + 3) * 16 + m];
      const float m0 = (q0 >= 0.0f) ? q0 * kmax : q0 * kmin;  // exact row max
      const float m1 = (q1 >= 0.0f) ? q1 * kmax : q1 * kmin;
      const float m2 = (q2 >= 0.0f) ? q2 * kmax : q2 * kmin;
      const float m3 = (q3 >= 0.0f) ? q3 * kmax : q3 * kmin;
      qs0 = q0 * LOG2E; nm0 = -m0 * LOG2E;
      qs1 = q1 * LOG2E; nm1 = -m1 * LOG2E;
      qs2 = q2 * LOG2E; nm2 = -m2 * LOG2E;
      qs3 = q3 * LOG2E; nm3 = -m3 * LOG2E;
    }

    v8f c0 = {}, c1 = {}, c2 = {}, c3 = {};
    #pragma unroll 2
    for (int ck = 0; ck < 64; ++ck) {
      const int kb = ck * 4 + 2 * h;
      const v2f kp = *(const v2f*)(kl + kb);   // ds_load_b64 (broadcast per half)
      const v2f vp = *(const v2f*)(vl + kb);   // ds_load_b64 (broadcast per half)
      v2f bm;                                  // branch-free B tile (pk_fma)
      bm.x = __fmaf_rn(sel_v, vp.x, sel_1);
      bm.y = __fmaf_rn(sel_v, vp.y, sel_1);

      // All logits into distinct temps first (no register reuse -> no WAR
      // hazard v_nops after v_exp), then all exps, then all WMMAs.
      const float t0x = __fmaf_rn(qs0, kp.x, nm0);
      const float t0y = __fmaf_rn(qs0, kp.y, nm0);
      const float t1x = __fmaf_rn(qs1, kp.x, nm1);
      const float t1y = __fmaf_rn(qs1, kp.y, nm1);
      const float t2x = __fmaf_rn(qs2, kp.x, nm2);
      const float t2y = __fmaf_rn(qs2, kp.y, nm2);
      const float t3x = __fmaf_rn(qs3, kp.x, nm3);
      const float t3y = __fmaf_rn(qs3, kp.y, nm3);

      v2f a0, a1, a2, a3;
      a0.x = exp2_fast(t0x);
      a0.y = exp2_fast(t0y);
      a1.x = exp2_fast(t1x);
      a1.y = exp2_fast(t1y);
      a2.x = exp2_fast(t2x);
      a2.y = exp2_fast(t2y);
      a3.x = exp2_fast(t3x);
      a3.y = exp2_fast(t3y);

      c0 = __builtin_amdgcn_wmma_f32_16x16x4_f32(false, a0, false, bm, (short)0, c0, false, false);
      c1 = __builtin_amdgcn_wmma_f32_16x16x4_f32(false, a1, false, bm, (short)0, c1, false, false);
      c2 = __builtin_amdgcn_wmma_f32_16x16x4_f32(false, a2, false, bm, (short)0, c2, false, false);
      c3 = __builtin_amdgcn_wmma_f32_16x16x4_f32(false, a3, false, bm, (short)0, c3, false, false);
    }

    EMIT_OUT(c0, g + 0)
    EMIT_OUT(c1, g + 1)
    EMIT_OUT(c2, g + 2)
    EMIT_OUT(c3, g + 3)
  }
}

extern "C" void kernel_launch(void* const* d_in, const int* in_sizes, int n_in,
                              void* d_out, int out_size, void* d_ws, size_t ws_size,
                              hipStream_t stream) {
  (void)n_in; (void)out_size; (void)d_ws; (void)ws_size;
  const float* q  = (const float*)d_in[0];
  const float* k  = (const float*)d_in[1];
  const float* v  = (const float*)d_in[2];
  const float* iw = (const float*)d_in[3];
  const float* ib = (const float*)d_in[4];
  const float* ow = (const float*)d_in[5];
  const float* ob = (const float*)d_in[6];
  float* out = (float*)d_out;

  const int L      = in_sizes[0] / NBATCH;     // 32768 (E=1)
  const int tiles  = (L / BSZ) * NBATCH;       // 128 * 32 = 4096 wave-tiles
  const int blocks = tiles / WPB;              // 512 workgroups of 256 threads

  block_attn_kernel<<<blocks, BSZ, 0, stream>>>(q, k, v, iw, ib, ow, ob, out);
}